// MinEuclideanDistBlock_7696581394831
// MI455X (gfx1250) — compile-verified
//
#include <hip/hip_runtime.h>
#include <math.h>

typedef __attribute__((ext_vector_type(2))) float v2f;
typedef __attribute__((ext_vector_type(8))) float v8f;

#define Bn 32
#define Cn 8
#define Ln 2048
#define Sn 64
#define Kn 256
#define Wn 1985
#define MT 32            // W rows per workgroup
#define SSTR 68          // LDS stride for staged shapelets (bank-conflict-free f32x2 reads)
#define DSTR 264         // LDS stride for distance-sum tile

// smem layout (floats):
//  xwin : Cn*96          = 768
//  x2s  : Cn*MT          = 256
//  red  : MT*8           = 256
//  dyn  : max(8*16*SSTR, MT*DSTR) = max(8704, 8448) = 8704   (union: swave during GEMM, dsum after)
#define SM_XWIN 0
#define SM_X2S  (SM_XWIN + Cn*96)
#define SM_RED  (SM_X2S + Cn*MT)
#define SM_DYN  (SM_RED + MT*8)
#define SM_TOT  (SM_DYN + 8704)

__global__ __launch_bounds__(256)
void shapelet_s2_kernel(const float* __restrict__ shp, float* __restrict__ s2) {
    int i = blockIdx.x * blockDim.x + threadIdx.x;   // 0..2047
    if (i < Cn * Kn) {
        const float* p = shp + i * Sn;
        float acc = 0.f;
        #pragma unroll
        for (int s = 0; s < Sn; ++s) { float v = p[s]; acc += v * v; }
        s2[i] = acc;
    }
}

// Load one 16-shapelet chunk (16 x 64 f32 = 4KB, contiguous) into 8 float4 regs.
__device__ __forceinline__ void load_chunk(float4 (&dst)[8], const float* __restrict__ shp,
                                           int c, int nt, int wave, int lane) {
    const float4* s = (const float4*)(shp + (c * Kn + wave * 32 + nt * 16) * Sn);
    #pragma unroll
    for (int j = 0; j < 8; ++j) dst[j] = s[j * 32 + lane];
}

// Store the register chunk into this wave's LDS staging area.
// g = j*32+lane  ->  row = 2*j + half, col = lm*4
__device__ __forceinline__ void store_chunk(const float4 (&src)[8], float* swave,
                                            int half, int lm) {
    #pragma unroll
    for (int j = 0; j < 8; ++j)
        *(float4*)&swave[(2 * j + half) * SSTR + (lm << 2)] = src[j];
}

__global__ __launch_bounds__(256)
void shapelet_main_kernel(const float* __restrict__ x,
                          const float* __restrict__ shp,
                          const float* __restrict__ s2g,
                          float* __restrict__ out) {
    __shared__ __align__(16) float smem[SM_TOT];
    float* xwin = smem + SM_XWIN;   // [Cn][96]
    float* x2s  = smem + SM_X2S;    // [Cn][MT]
    float* red  = smem + SM_RED;    // [MT][8]
    float* dyn  = smem + SM_DYN;    // swave [8][16][SSTR]  /  dsum [MT][DSTR]

    const int tid  = threadIdx.x;
    const int wave = tid >> 5;
    const int lane = tid & 31;
    const int half = lane >> 4;     // which half-wave (selects K pair / M rows 8..15)
    const int lm   = lane & 15;
    const int w0   = blockIdx.x * MT;
    const int b    = blockIdx.y;

    // ---- stage sliding-window source data for all 8 channels -------------
    for (int i = tid; i < Cn * 96; i += 256) {
        int c = i / 96, off = i % 96;
        int idx = w0 + off;
        xwin[c * 96 + off] = (idx < Ln) ? x[(b * Cn + c) * Ln + idx] : 0.f;
    }
    __syncthreads();

    // ---- per-row window norms: 256 threads <-> 8 channels x 32 rows ------
    {
        int c = tid >> 5, row = tid & 31;
        float acc = 0.f;
        #pragma unroll
        for (int s = 0; s < Sn; ++s) { float v = xwin[c * 96 + row + s]; acc += v * v; }
        x2s[c * MT + row] = acc;
    }
    __syncthreads();

    float* swave = dyn + wave * (16 * SSTR);   // this wave's private staging area
    const int kk0 = wave * 32 + lm;
    const int kk1 = kk0 + 16;

    v8f d00 = {}, d01 = {}, d10 = {}, d11 = {};   // [mtile][ntile] distance-sum accumulators

    // One chunk = 16 shapelets of one channel (ch = c*2 + nt).  32 WMMA per chunk.
    auto compute = [&](int c, float s2v, v8f& dA, v8f& dB) {
        v8f accA = {}, accB = {};              // rows 0-15 / rows 16-31
        #pragma unroll
        for (int s0 = 0; s0 < Sn; s0 += 4) {
            int sb = s0 + 2 * half;            // K pair {0,1} or {2,3}
            v2f A0, A1;
            A0.x = xwin[c * 96 + lm + sb];
            A0.y = xwin[c * 96 + lm + sb + 1];
            A1.x = xwin[c * 96 + 16 + lm + sb];
            A1.y = xwin[c * 96 + 16 + lm + sb + 1];
            v2f Bm = *(const v2f*)&swave[lm * SSTR + sb];   // 8B aligned, conflict-free
            accA = __builtin_amdgcn_wmma_f32_16x16x4_f32(false, A0, false, Bm,
                                                         (short)0, accA, false, false);
            accB = __builtin_amdgcn_wmma_f32_16x16x4_f32(false, A1, false, Bm,
                                                         (short)0, accB, false, false);
        }
        #pragma unroll
        for (int r = 0; r < 8; ++r) {
            int r0 = r + 8 * half;
            float xa = x2s[c * MT + r0];
            float xb = x2s[c * MT + 16 + r0];
            // args clamped >= 1e-12: single-instruction v_sqrt_f32 (TRANS pipe,
            // co-executes with WMMA) instead of the 10-op correctly-rounded expansion
            dA[r] += __builtin_amdgcn_sqrtf(fmaxf(xa + s2v - 2.f * accA[r], 1e-12f));
            dB[r] += __builtin_amdgcn_sqrtf(fmaxf(xb + s2v - 2.f * accB[r], 1e-12f));
        }
    };

    // ---- software-pipelined chunk loop: global loads for chunk i+1 -------
    // ---- overlap the 32-WMMA compute phase of chunk i --------------------
    float4 P[8], Q[8];
    load_chunk(P, shp, 0, 0, wave, lane);
    for (int ch = 0; ch < 16; ch += 2) {
        int c = ch >> 1;
        store_chunk(P, swave, half, lm);               // waits loadcnt once, then 8 ds_store
        load_chunk(Q, shp, c, 1, wave, lane);          // prefetch odd chunk (overlaps compute)
        compute(c, s2g[c * Kn + kk0], d00, d10);       // even chunk: columns kk0
        store_chunk(Q, swave, half, lm);               // DS in-order: safe vs compute's ds_loads
        if (ch < 14) load_chunk(P, shp, c + 1, 0, wave, lane);  // prefetch next even chunk
        compute(c, s2g[c * Kn + kk1], d01, d11);       // odd chunk: columns kk1
    }

    // all waves done reading their swave regions before dsum aliases them
    __syncthreads();
    float* dsum = dyn;                              // [MT][DSTR]
    #pragma unroll
    for (int r = 0; r < 8; ++r) {
        int r0 = r + 8 * half;
        dsum[r0 * DSTR + kk0]        = d00[r];
        dsum[r0 * DSTR + kk1]        = d01[r];
        dsum[(16 + r0) * DSTR + kk0] = d10[r];
        dsum[(16 + r0) * DSTR + kk1] = d11[r];
    }
    __syncthreads();

    // ---- min over K = 256 per row: 8 threads per row, then 8->1 ----------
    {
        int row = tid >> 3, g = tid & 7;
        float m = 3.4e38f;
        #pragma unroll
        for (int j = 0; j < 32; ++j) m = fminf(m, dsum[row * DSTR + g * 32 + j]);
        red[row * 8 + g] = m;
    }
    __syncthreads();
    if (tid < MT) {
        float m = 3.4e38f;
        #pragma unroll
        for (int g = 0; g < 8; ++g) m = fminf(m, red[tid * 8 + g]);
        int w = w0 + tid;
        if (w < Wn) out[b * Wn + w] = m;
    }
}

extern "C" void kernel_launch(void* const* d_in, const int* in_sizes, int n_in,
                              void* d_out, int out_size, void* d_ws, size_t ws_size,
                              hipStream_t stream) {
    const float* x   = (const float*)d_in[0];   // (32, 8, 2048) f32
    const float* shp = (const float*)d_in[1];   // (8, 256, 64) f32
    float* out = (float*)d_out;                 // (32, 1, 1985) f32
    float* s2  = (float*)d_ws;                  // 2048 floats of shapelet norms

    shapelet_s2_kernel<<<dim3((Cn * Kn + 255) / 256), dim3(256), 0, stream>>>(shp, s2);
    shapelet_main_kernel<<<dim3((Wn + MT - 1) / MT, Bn), dim3(256), 0, stream>>>(x, shp, s2, out);
}